// EMD_Loss_79800492360030
// MI455X (gfx1250) — compile-verified
//
#include <hip/hip_runtime.h>

// Sinkhorn divergence (geomloss 'sinkhorn', p=2, blur=0.01, scaling=0.9, debiased)
// B=8, N=M=2048, D=3.  Cost matrices recomputed on the fly from LDS-resident
// point clouds; x.y^T tiles via V_WMMA_F32_16X16X4_F32.  The unused 4th K slot
// carries the per-column bias t_j (A[m][3]=1, B[3][n]=t_n), so the WMMA output
// is directly the logsumexp argument.  Streaming base-2 logsumexp with raw
// v_exp_f32, one transcendental per matrix element; running max via cndmask
// (reusing the compare) to avoid canonicalize+max lowering of fmaxf.

#define NPTS 2048
#define BATCH 8
#define LOG2E 1.4426950408889634f
#define LN2 0.6931471805599453f
#define LOG2_N 11.0f /* log2(2048) */

typedef __attribute__((ext_vector_type(2))) float v2f;
typedef __attribute__((ext_vector_type(8))) float v8f;

// ---------------------------------------------------------------------------
// init: pack points as [B][N][4] (pad 4th comp = 0), 0.5*|p|^2, zero potentials
// ---------------------------------------------------------------------------
__global__ __launch_bounds__(256) void sink_init_kernel(
    const float* __restrict__ p1, const float* __restrict__ p2,
    float* __restrict__ x4, float* __restrict__ y4,
    float* __restrict__ sqx, float* __restrict__ sqy,
    float* __restrict__ pots0) {
  int i = blockIdx.x * blockDim.x + threadIdx.x;  // 0 .. B*N-1
  const int BN = BATCH * NPTS;
  if (i >= BN) return;
  {
    float a0 = p1[i * 3 + 0], a1 = p1[i * 3 + 1], a2 = p1[i * 3 + 2];
    x4[i * 4 + 0] = a0; x4[i * 4 + 1] = a1;
    x4[i * 4 + 2] = a2; x4[i * 4 + 3] = 0.0f;
    sqx[i] = 0.5f * (a0 * a0 + a1 * a1 + a2 * a2);
  }
  {
    float b0 = p2[i * 3 + 0], b1 = p2[i * 3 + 1], b2 = p2[i * 3 + 2];
    y4[i * 4 + 0] = b0; y4[i * 4 + 1] = b1;
    y4[i * 4 + 2] = b2; y4[i * 4 + 3] = 0.0f;
    sqy[i] = 0.5f * (b0 * b0 + b1 * b1 + b2 * b2);
  }
  pots0[0 * BN + i] = 0.0f;  // f
  pots0[1 * BN + i] = 0.0f;  // g
  pots0[2 * BN + i] = 0.0f;  // fx
  pots0[3 * BN + i] = 0.0f;  // gy
}

// ---------------------------------------------------------------------------
// One annealing step: all four softmins.
//   q=0: f  <- softmin(Cxy, g)    rows=x cols=y
//   q=1: g  <- softmin(Cyx, f)    rows=y cols=x
//   q=2: fx <- softmin(Cxx, fx)   rows=x cols=x
//   q=3: gy <- softmin(Cyy, gy)   rows=y cols=y
// softmin_i = 0.5|row_i|^2 - eps*ln2 * log2sumexp2_j( t_j + (row_i . col_j)*scale )
//   t_j = (h_j - 0.5|col_j|^2)*scale - log2(M),  scale = log2(e)/eps
// The WMMA computes the full argument: A = [x*scale | 1], B = [y | t]^T, K=4.
// mode 0: out = 0.5*(out_old + softmin);  mode 1: out = softmin (final)
// ---------------------------------------------------------------------------
__global__ __launch_bounds__(256) void sinkhorn_softmin_kernel(
    const float* __restrict__ x4, const float* __restrict__ y4,
    const float* __restrict__ sqx, const float* __restrict__ sqy,
    const float* __restrict__ potsOld, float* __restrict__ potsNew,
    float eps, int mode) {
  __shared__ float colsLds[NPTS][4];  // 32 KB: {c0, c1, c2, t_j}

  const int b = blockIdx.z;
  const int q = blockIdx.y;
  const int chunk = blockIdx.x;  // 0..15 -> 128 rows each
  const int tid = threadIdx.x;
  const int lane = tid & 31;
  const int wave = tid >> 5;  // 0..7
  const int BN = BATCH * NPTS;

  const float* rowSrc = (q & 1) ? y4 : x4;
  const float* colSrc = (q == 0 || q == 3) ? y4 : x4;
  const float* sqRow = (q & 1) ? sqy : sqx;
  const int hIdx = (q < 2) ? (q ^ 1) : q;

  const float scale = LOG2E / eps;
  const float* hVec = potsOld + (size_t)hIdx * BN + (size_t)b * NPTS;

  // Fill LDS with column points + per-column bias t_j (base-2) in slot 3.
  for (int j = tid; j < NPTS; j += 256) {
    const float* cp = colSrc + ((size_t)b * NPTS + j) * 4;
    float c0 = cp[0], c1 = cp[1], c2 = cp[2];
    colsLds[j][0] = c0;
    colsLds[j][1] = c1;
    colsLds[j][2] = c2;
    float sq = 0.5f * (c0 * c0 + c1 * c1 + c2 * c2);
    colsLds[j][3] = (hVec[j] - sq) * scale - LOG2_N;
  }
  __syncthreads();

  // A tile (16x4 f32, ISA layout): lane holds row rowBase+(lane&15),
  // K components {k0, k0+1}, k0 = 2*(lane>>4).  Components 0..2 carry
  // x*scale, component 3 carries 1.0 (bias row).
  const int rowBase = chunk * 128 + wave * 16;
  const int rIn16 = lane & 15;
  const int k0 = (lane >> 4) * 2;
  const float* rp = rowSrc + ((size_t)b * NPTS + rowBase + rIn16) * 4 + k0;
  v2f aTile;
  aTile.x = rp[0] * scale;
  aTile.y = (k0 == 2) ? 1.0f : rp[1] * scale;

  // Streaming log2-sum-exp2 accumulators. D layout: VGPR r, lane-half H
  // holds row (r + 8*H), column (n0 + lane&15).
  float mAcc[8], sAcc[8];
#pragma unroll
  for (int r = 0; r < 8; ++r) {
    mAcc[r] = -__builtin_inff();
    sAcc[r] = 0.0f;
  }

  v8f czero = {};
#pragma unroll 4
  for (int n0 = 0; n0 < NPTS; n0 += 16) {
    // B tile (4x16 f32): lane holds col n0+(lane&15), K comps {k0, k0+1}.
    // Upper half loads {c2, t_j}.
    v2f bTile;
    bTile.x = colsLds[n0 + rIn16][k0];
    bTile.y = colsLds[n0 + rIn16][k0 + 1];

    v8f d = __builtin_amdgcn_wmma_f32_16x16x4_f32(
        /*neg_a=*/false, aTile, /*neg_b=*/false, bTile,
        /*c_mod=*/(short)0, czero, /*reuse_a=*/false, /*reuse_b=*/false);

#pragma unroll
    for (int r = 0; r < 8; ++r) {
      float aVal = d[r];
      float mOld = mAcc[r];
      bool gt = aVal > mOld;
      // one raw v_exp_f32 per element; huge negative args flush to 0 (wanted)
      float e = __builtin_amdgcn_exp2f(-__builtin_fabsf(aVal - mOld));
      float emul = gt ? e : 1.0f;
      float eadd = gt ? 1.0f : e;
      sAcc[r] = __builtin_fmaf(sAcc[r], emul, eadd);
      mAcc[r] = gt ? aVal : mOld;  // running max via cndmask (no canonicalize)
    }
  }

  // Cross-lane merge within each 16-lane half, then write per-row results.
#pragma unroll
  for (int r = 0; r < 8; ++r) {
    float m = mAcc[r], s = sAcc[r];
#pragma unroll
    for (int msk = 1; msk <= 8; msk <<= 1) {
      float om = __shfl_xor(m, msk, 32);
      float os = __shfl_xor(s, msk, 32);
      bool gt = m > om;
      float M = gt ? m : om;
      s = s * __builtin_amdgcn_exp2f(m - M) +
          os * __builtin_amdgcn_exp2f(om - M);
      m = M;
    }
    if (rIn16 == 0) {
      int row = rowBase + r + ((lane >> 4) << 3);
      float L = m + __builtin_amdgcn_logf(s);  // v_log_f32 = log2
      float sm = sqRow[b * NPTS + row] - eps * LN2 * L;
      size_t oidx = (size_t)q * BN + (size_t)b * NPTS + row;
      if (mode == 0)
        potsNew[oidx] = 0.5f * (potsOld[oidx] + sm);
      else
        potsNew[oidx] = sm;
    }
  }
}

// ---------------------------------------------------------------------------
// Final reduction: mean_b[ mean_i(f - fx) + mean_j(g - gy) ]
// ---------------------------------------------------------------------------
__global__ __launch_bounds__(256) void sink_reduce_kernel(
    const float* __restrict__ res, float* __restrict__ out) {
  const int BN = BATCH * NPTS;
  float acc = 0.0f;
  for (int i = threadIdx.x; i < BN; i += 256) {
    acc += (res[0 * BN + i] - res[2 * BN + i]) +
           (res[1 * BN + i] - res[3 * BN + i]);
  }
  __shared__ float red[256];
  red[threadIdx.x] = acc;
  __syncthreads();
  for (int s = 128; s > 0; s >>= 1) {
    if (threadIdx.x < (unsigned)s) red[threadIdx.x] += red[threadIdx.x + s];
    __syncthreads();
  }
  if (threadIdx.x == 0) out[0] = red[0] / (float)BN;
}

// ---------------------------------------------------------------------------
extern "C" void kernel_launch(void* const* d_in, const int* in_sizes, int n_in,
                              void* d_out, int out_size, void* d_ws,
                              size_t ws_size, hipStream_t stream) {
  (void)in_sizes; (void)n_in; (void)out_size; (void)ws_size;
  const float* p1 = (const float*)d_in[0];
  const float* p2 = (const float*)d_in[1];
  float* out = (float*)d_out;

  const size_t BN = (size_t)BATCH * NPTS;
  float* ws = (float*)d_ws;
  float* x4 = ws;                 // BN*4
  float* y4 = x4 + BN * 4;        // BN*4
  float* sqx = y4 + BN * 4;       // BN
  float* sqy = sqx + BN;          // BN
  float* pots0 = sqy + BN;        // 4*BN
  float* pots1 = pots0 + 4 * BN;  // 4*BN

  sink_init_kernel<<<(int)((BN + 255) / 256), 256, 0, stream>>>(
      p1, p2, x4, y4, sqx, sqy, pots0);

  dim3 grid(16, 4, BATCH);

  // eps annealing: s = 8.0 * 0.9^k while s > 0.01, eps = s^2; then blur^2.
  int p = 0;
  double s = 8.0;
  while (s > 0.01) {
    float eps = (float)(s * s);
    sinkhorn_softmin_kernel<<<grid, 256, 0, stream>>>(
        x4, y4, sqx, sqy, p ? pots1 : pots0, p ? pots0 : pots1, eps, 0);
    p ^= 1;
    s *= 0.9;
  }
  const float epsT = (float)(0.01 * 0.01);
  // last scan step at blur^2
  sinkhorn_softmin_kernel<<<grid, 256, 0, stream>>>(
      x4, y4, sqx, sqy, p ? pots1 : pots0, p ? pots0 : pots1, epsT, 0);
  p ^= 1;
  // final differentiable extrapolation (raw softmin values)
  float* cur = p ? pots1 : pots0;
  float* res = p ? pots0 : pots1;
  sinkhorn_softmin_kernel<<<grid, 256, 0, stream>>>(
      x4, y4, sqx, sqy, cur, res, epsT, 1);

  sink_reduce_kernel<<<1, 256, 0, stream>>>(res, out);
}